// HolographicTTEmbedding_34505767256773
// MI455X (gfx1250) — compile-verified
//
#include <hip/hip_runtime.h>

typedef __attribute__((ext_vector_type(2))) float v2f;
typedef __attribute__((ext_vector_type(8))) float v8f;

#define TT_V1   225
#define TT_V2   224
#define TT_RANK 16
#define TT_D1   32
#define TT_D2   32

// One wave handles one token per iteration:
//   out(32x32) = clip(c1*cos(phase))^T (32x16)  x  clip(c2) (16x32), * 0.25
// computed as 4 16x16 tiles, each via 4 chained V_WMMA_F32_16X16X4_F32.
__global__ __launch_bounds__(256) void
holo_tt_embed_wmma(const int*   __restrict__ ids,
                   const float* __restrict__ core1,   // [V1][RANK][D1]
                   const float* __restrict__ core2,   // [V2][RANK][D2]
                   const float* __restrict__ phase,   // [RANK]
                   float*       __restrict__ out,     // [ntok][1024]
                   int ntok)
{
    const int lane = threadIdx.x & 31;
    const int hi   = lane >> 4;     // 0: lanes 0-15, 1: lanes 16-31
    const int lo   = lane & 15;

    const int wavesPerBlock = blockDim.x >> 5;
    const int w = blockIdx.x * wavesPerBlock + (threadIdx.x >> 5);
    const int W = gridDim.x * wavesPerBlock;

    // Loop-invariant: the 8 cos(phase[r]) values this lane ever multiplies.
    // A-matrix 16x4 f32 layout: lane<16 holds K=0,1 ; lane>=16 holds K=2,3,
    // so per k-step s and VGPR j:  r = 4*s + 2*hi + j.
    float cosv[4][2];
#pragma unroll
    for (int s = 0; s < 4; ++s)
#pragma unroll
        for (int j = 0; j < 2; ++j)
            cosv[s][j] = cosf(phase[4 * s + 2 * hi + j]);

    for (int t = w; t < ntok; t += W) {
        const int id = ids[t];                       // 0 <= id < 50257
        int i1 = id / TT_V2;
        i1 = (i1 > TT_V1 - 1) ? (TT_V1 - 1) : i1;    // clip (upper only; id >= 0)
        const int i2 = id % TT_V2;                   // already in [0, V2)

        const float* __restrict__ c1p = core1 + (size_t)i1 * (TT_RANK * TT_D1);
        const float* __restrict__ c2p = core2 + (size_t)i2 * (TT_RANK * TT_D2);

        // Gather A (c1^T) and B (c2) fragments for all 4 k-steps, both halves.
        // a[dh][s][j]: element (M = dh*16+lo rows of out, K = 4s+2hi+j)
        // b[fh][s][j]: element (K = 4s+2hi+j, N = fh*16+lo)
        float a[2][4][2], b[2][4][2];
#pragma unroll
        for (int s = 0; s < 4; ++s) {
#pragma unroll
            for (int j = 0; j < 2; ++j) {
                const int r = 4 * s + 2 * hi + j;
#pragma unroll
                for (int h = 0; h < 2; ++h) {
                    float ra = c1p[r * TT_D1 + h * 16 + lo] * cosv[s][j];
                    ra = fminf(fmaxf(ra, -10.0f), 10.0f);
                    a[h][s][j] = ra;
                    float rb = c2p[r * TT_D2 + h * 16 + lo];
                    rb = fminf(fmaxf(rb, -10.0f), 10.0f);
                    b[h][s][j] = rb;
                }
            }
        }

        // 4 output tiles, K=16 contraction via 4 x V_WMMA_F32_16X16X4_F32 each.
        v8f acc[2][2];
#pragma unroll
        for (int dh = 0; dh < 2; ++dh) {
#pragma unroll
            for (int fh = 0; fh < 2; ++fh) {
                v8f c = {};
#pragma unroll
                for (int s = 0; s < 4; ++s) {
                    v2f av; av.x = a[dh][s][0]; av.y = a[dh][s][1];
                    v2f bv; bv.x = b[fh][s][0]; bv.y = b[fh][s][1];
                    c = __builtin_amdgcn_wmma_f32_16x16x4_f32(
                            /*neg_a=*/false, av, /*neg_b=*/false, bv,
                            /*c_mod=*/(short)0, c,
                            /*reuse_a=*/false, /*reuse_b=*/false);
                }
                acc[dh][fh] = c;
            }
        }

        // Store: C/D layout -> element (M = j + 8*hi, N = lo) in VGPR j.
        // out[t][ (dh*16 + M) * 32 + fh*16 + N ] = acc * 1/sqrt(RANK).
        // Non-temporal: 134 MB streaming output, keep cores cache-resident.
        float* __restrict__ op = out + (size_t)t * (TT_D1 * TT_D2);
#pragma unroll
        for (int dh = 0; dh < 2; ++dh) {
#pragma unroll
            for (int fh = 0; fh < 2; ++fh) {
#pragma unroll
                for (int j = 0; j < 8; ++j) {
                    const int d = dh * 16 + j + 8 * hi;
                    const int f = fh * 16 + lo;
                    __builtin_nontemporal_store(acc[dh][fh][j] * 0.25f,
                                                &op[d * 32 + f]);
                }
            }
        }
    }
}

extern "C" void kernel_launch(void* const* d_in, const int* in_sizes, int n_in,
                              void* d_out, int out_size, void* d_ws, size_t ws_size,
                              hipStream_t stream) {
    const int*   ids   = (const int*)  d_in[0];   // input_ids (int32 under default JAX)
    const float* core1 = (const float*)d_in[1];   // (225,1,16,32)
    const float* core2 = (const float*)d_in[2];   // (224,16,1,32)
    const float* phase = (const float*)d_in[3];   // (16,)
    float* out = (float*)d_out;

    const int ntok = in_sizes[0];                 // B*L = 32768

    dim3 block(256);                              // 8 waves / workgroup
    dim3 grid(512);                               // 4096 waves -> 8 tokens each
    hipLaunchKernelGGL(holo_tt_embed_wmma, grid, block, 0, stream,
                       ids, core1, core2, phase, out, ntok);
}